// Blur_29308856828607
// MI455X (gfx1250) — compile-verified
//
#include <hip/hip_runtime.h>
#include <hip/hip_bf16.h>
#include <math.h>
#include <stdint.h>

// float4 that may be stored at 4-byte alignment (output row stride 511 is odd)
typedef float v4f __attribute__((ext_vector_type(4), aligned(4)));

#define IN_H 256
#define IN_W 256
#define ROWS 16          // input rows per workgroup
#define LDSW 264         // LDS row stride in floats (1056 B, 16B aligned)
#define DCOL 4           // first data column in an LDS row (byte offset 16)

// CDNA5 async global->LDS copy (ASYNCcnt-tracked, per-lane 16B, GV addressing)
__device__ __forceinline__ void async_load16(uint32_t lds_off, const float* g) {
  asm volatile("global_load_async_to_lds_b128 %0, %1, off"
               :: "v"(lds_off), "v"(g)
               : "memory");
}

__device__ __forceinline__ void wait_async0() {
#if __has_builtin(__builtin_amdgcn_s_wait_asynccnt)
  __builtin_amdgcn_s_wait_asynccnt(0);
#else
  asm volatile("s_wait_asynccnt 0" ::: "memory");
#endif
}

__global__ __launch_bounds__(256) void blur_up2_kernel(
    const float* __restrict__ x, const float* __restrict__ ker,
    float* __restrict__ out, int ow) {
  __shared__ __align__(16) float sh[(ROWS + 2) * LDSW];

  const int tid   = threadIdx.x;
  const int r0    = blockIdx.x * ROWS;   // first input row of this tile
  const int plane = blockIdx.y;          // n*C + c

  const float* xp = x + (size_t)plane * (IN_H * IN_W);

  // zero side-halo columns (left of col 0 / right of col 255 -> outside plane)
  if (tid < ROWS + 2) {
    sh[tid * LDSW + (DCOL - 1)]    = 0.0f;
    sh[tid * LDSW + (DCOL + IN_W)] = 0.0f;
  }

  // async-copy 18 rows x 256 floats to LDS; out-of-plane rows are zero-filled
  for (int l = tid; l < (ROWS + 2) * (IN_W / 4); l += 256) {
    const int row = l >> 6;           // IN_W/4 == 64 b128 chunks per row
    const int c   = (l & 63) * 4;
    const int gr  = r0 - 1 + row;
    float* dst = &sh[row * LDSW + DCOL + c];
    if ((unsigned)gr < (unsigned)IN_H) {
      async_load16((uint32_t)(uintptr_t)dst, xp + gr * IN_W + c);
    } else {
      v4f z = 0.0f;
      *(v4f*)dst = z;                  // 16B-aligned LDS store
    }
  }
  wait_async0();
  __syncthreads();

  // flipped 4x4 taps (symmetric filter, flip kept for faithfulness)
  const float k00 = ker[15], k01 = ker[14], k02 = ker[13], k03 = ker[12];
  const float k10 = ker[11], k11 = ker[10], k12 = ker[9],  k13 = ker[8];
  const float k20 = ker[7],  k21 = ker[6],  k22 = ker[5],  k23 = ker[4];
  const float k30 = ker[3],  k31 = ker[2],  k32 = ker[1],  k33 = ker[0];

  float* op = out + (size_t)plane * ow * ow;

  // 16 rows x 128 input-col-pairs per tile; each thread does 8 pairs.
  // A pair (ci, ci+1) produces a 2-row x 4-col output block at (2m, 4*pc).
  for (int k = 0; k < 8; ++k) {
    const int p  = k * 256 + tid;
    const int ri = p >> 7;             // 0..15
    const int pc = p & 127;            // 0..127  (adjacent lanes -> adjacent pc)
    const int base = (ri + 1) * LDSW + DCOL + 2 * pc;

    const float A0 = sh[base - LDSW - 1], A1 = sh[base - LDSW],
                A2 = sh[base - LDSW + 1], A3 = sh[base - LDSW + 2];
    const float B0 = sh[base - 1],        B1 = sh[base],
                B2 = sh[base + 1],        B3 = sh[base + 2];
    const float C0 = sh[base + LDSW - 1], C1 = sh[base + LDSW],
                C2 = sh[base + LDSW + 1], C3 = sh[base + LDSW + 2];

    v4f top, bot;
    top.x = k00*A0 + k02*A1 + k20*B0 + k22*B1;   // (even row, even col) of ci
    top.y = k01*A1 + k03*A2 + k21*B1 + k23*B2;   // (even, odd)  of ci
    top.z = k00*A1 + k02*A2 + k20*B1 + k22*B2;   // (even, even) of ci+1
    top.w = k01*A2 + k03*A3 + k21*B2 + k23*B3;   // (even, odd)  of ci+1
    bot.x = k10*B0 + k12*B1 + k30*C0 + k32*C1;   // (odd row) variants
    bot.y = k11*B1 + k13*B2 + k31*C1 + k33*C2;
    bot.z = k10*B1 + k12*B2 + k30*C1 + k32*C2;
    bot.w = k11*B2 + k13*B3 + k31*C2 + k33*C3;

    const int orow = 2 * (r0 + ri);
    const int ocol = 4 * pc;
    float* o0 = op + (size_t)orow * ow + ocol;
    const bool botv = (orow + 1) < ow;           // clip last odd row (ow==511)

    if (ocol + 4 <= ow) {
      __builtin_nontemporal_store(top, (v4f*)o0);
      if (botv) __builtin_nontemporal_store(bot, (v4f*)(o0 + ow));
    } else {                                      // ragged right edge (ow==511)
      const int nrem = ow - ocol;                 // 1..3
      const float t[4] = {top.x, top.y, top.z, top.w};
      const float b[4] = {bot.x, bot.y, bot.z, bot.w};
      for (int j = 0; j < nrem; ++j) {
        __builtin_nontemporal_store(t[j], o0 + j);
        if (botv) __builtin_nontemporal_store(b[j], o0 + ow + j);
      }
    }
  }
}

extern "C" void kernel_launch(void* const* d_in, const int* in_sizes, int n_in,
                              void* d_out, int out_size, void* d_ws, size_t ws_size,
                              hipStream_t stream) {
  (void)n_in; (void)d_ws; (void)ws_size;
  const float* x   = (const float*)d_in[0];   // [4,64,256,256] f32
  const float* ker = (const float*)d_in[1];   // [4,4] f32
  float* out = (float*)d_out;

  const int planes = in_sizes[0] / (IN_H * IN_W);                 // N*C = 256
  const int ow = (int)(sqrt((double)out_size / (double)planes) + 0.5);  // 511 (or 512)

  dim3 grid(IN_H / ROWS, planes);   // 16 x 256 workgroups
  blur_up2_kernel<<<grid, 256, 0, stream>>>(x, ker, out, ow);
}